// MF_FractalNet_20796231647839
// MI455X (gfx1250) — compile-verified
//
#include <hip/hip_runtime.h>
#include <hip/hip_bf16.h>

#define N_NODES 100000
#define G_NODES 50000
#define N_EDGES 800000
#define HID 64
#define F_IN 32
#define MAX_DEG 20
#define NUM_GRAPHS 128
#define LN_EPS 1e-5f

typedef __attribute__((ext_vector_type(2))) float v2f;
typedef __attribute__((ext_vector_type(8))) float v8f;

__device__ __forceinline__ v8f wmma4(v2f a, v2f b, v8f c) {
    // V_WMMA_F32_16X16X4_F32 : D = A(16x4) x B(4x16) + C(16x16)
    return __builtin_amdgcn_wmma_f32_16x16x4_f32(false, a, false, b, (short)0, c,
                                                 false, false);
}

// Async global->LDS copy, 16B per lane, tracked by ASYNCcnt (CDNA5 TDM-lite path).
__device__ __forceinline__ void async_copy_b128(uint32_t lds_byte_off, const void* gptr) {
    asm volatile("global_load_async_to_lds_b128 %0, %1, off"
                 :: "v"(lds_byte_off), "v"(gptr) : "memory");
}
__device__ __forceinline__ void wait_asynccnt0() {
    asm volatile("s_wait_asynccnt 0x0" ::: "memory");
}

// ---------------------------------------------------------------- utilities
__global__ void k_zero(float* p, int n) {
    int i = blockIdx.x * blockDim.x + threadIdx.x;
    if (i < n) p[i] = 0.0f;
}
__global__ void k_zero_i(int* p, int n) {
    int i = blockIdx.x * blockDim.x + threadIdx.x;
    if (i < n) p[i] = 0;
}

// ---------------------------------------------------------------- embedding
// h = x @ emb_w + emb_b ; also copy into h0 (residual). One wave per 16 rows.
__global__ void k_embed(const float* __restrict__ x, const float* __restrict__ w,
                        const float* __restrict__ bias, float* __restrict__ h,
                        float* __restrict__ h0, int n) {
    int tile = blockIdx.x;
    int lane = threadIdx.x;      // 0..31
    int row = lane & 15;         // M within tile
    int khalf = lane >> 4;       // which K pair
    int gr = tile * 16 + row;
    bool fullTile = (tile * 16 + 16) <= n;
    bool valid = gr < n;

    v2f afrag[8];
#pragma unroll
    for (int ks = 0; ks < 8; ++ks) {
        int kk = ks * 4 + 2 * khalf;
        if (valid) {
            afrag[ks].x = x[gr * F_IN + kk];
            afrag[ks].y = x[gr * F_IN + kk + 1];
        } else {
            afrag[ks].x = 0.0f; afrag[ks].y = 0.0f;
        }
    }
#pragma unroll
    for (int n0 = 0; n0 < HID; n0 += 16) {
        v8f acc = {};
#pragma unroll
        for (int ks = 0; ks < 8; ++ks) {
            int kk = ks * 4 + 2 * khalf;
            v2f b;
            b.x = w[kk * HID + n0 + row];
            b.y = w[(kk + 1) * HID + n0 + row];
            acc = wmma4(afrag[ks], b, acc);
        }
        int col = n0 + row;
        float bb = bias[col];
        if (fullTile) {
#pragma unroll
            for (int r = 0; r < 8; ++r) {
                int g = tile * 16 + r + 8 * khalf;
                float v = acc[r] + bb;
                h[g * HID + col] = v;
                h0[g * HID + col] = v;
            }
        } else {
#pragma unroll
            for (int r = 0; r < 8; ++r) {
                int g = tile * 16 + r + 8 * khalf;
                if (g < n) {
                    float v = acc[r] + bb;
                    h[g * HID + col] = v;
                    h0[g * HID + col] = v;
                }
            }
        }
    }
}

// ---------------------------------------------------------------- graph prep
__global__ void k_deg(const int* __restrict__ dst, int* __restrict__ deg, int e) {
    int i = blockIdx.x * blockDim.x + threadIdx.x;
    if (i < e) atomicAdd(&deg[dst[i]], 1);
}

__global__ void k_scatter(const int* __restrict__ src, const int* __restrict__ dst,
                          const float* __restrict__ h, float* __restrict__ hsum, int e) {
    int g = blockIdx.x * blockDim.x + threadIdx.x;
    int ei = g >> 6, c = g & 63;
    if (ei < e) {
        unsafeAtomicAdd(&hsum[dst[ei] * HID + c], h[src[ei] * HID + c]);
    }
}

// meta layout (ints): counts[0..20] offsets[21..41] cursor[42..62] toff[63..84]
__global__ void k_hist(const int* __restrict__ deg, int* __restrict__ meta, int n) {
    int i = blockIdx.x * blockDim.x + threadIdx.x;
    if (i >= n) return;
    int d = deg[i]; if (d > MAX_DEG) d = MAX_DEG;
    atomicAdd(&meta[d], 1);
}

__global__ void k_scan(int* meta) {
    if (threadIdx.x == 0 && blockIdx.x == 0) {
        int* counts = meta; int* offsets = meta + 21;
        int* cursor = meta + 42; int* toff = meta + 63;
        int off = 0, tf = 0;
        for (int k = 0; k <= MAX_DEG; ++k) {
            offsets[k] = off; toff[k] = tf; cursor[k] = 0;
            off += counts[k];
            tf += (counts[k] + 15) >> 4;
        }
        toff[21] = tf;
    }
}

__global__ void k_permfill(const int* __restrict__ deg, int* __restrict__ perm,
                           int* __restrict__ meta, int n) {
    int i = blockIdx.x * blockDim.x + threadIdx.x;
    if (i >= n) return;
    int d = deg[i]; if (d > MAX_DEG) d = MAX_DEG;
    int pos = atomicAdd(&meta[42 + d], 1);
    perm[meta[21 + d] + pos] = i;
}

// ---------------------------------------------------------------- MFConv GEMM
// One wave per 16-row tile of one degree bucket.
// out = hsum_tile @ wl + hin_tile @ wr + bl, same slice for every k (relu if requested)
// bucket 0 (deg==0): passthrough of hin (+relu) — PyG update-mask semantics.
__global__ void k_conv(const float* __restrict__ hsum, const float* __restrict__ hin,
                       float* __restrict__ hout, const int* __restrict__ perm,
                       const int* __restrict__ meta, const float* __restrict__ wl,
                       const float* __restrict__ bl, const float* __restrict__ wr,
                       int do_relu) {
    __shared__ __align__(16) float sH[16 * HID];
    __shared__ __align__(16) float sX[16 * HID];
    const int* counts = meta;
    const int* offsets = meta + 21;
    const int* toff = meta + 63;

    int t = blockIdx.x;
    if (t >= toff[21]) return;
    int k = 0;
    while (!(t >= toff[k] && t < toff[k + 1])) ++k;   // <=21 scalar iters
    int tileIn = t - toff[k];
    int rowbase = offsets[k] + tileIn * 16;
    int cnt = counts[k];
    int lane = threadIdx.x;
    bool full = (tileIn * 16 + 16) <= cnt;

    if (k == 0) {                     // deg==0 -> not updated: keep h (relu'd)
        for (int r = 0; r < 16; ++r) {
            if (tileIn * 16 + r >= cnt) break;
            int node = perm[rowbase + r];
#pragma unroll
            for (int c = lane; c < HID; c += 32) {
                float v = hin[node * HID + c];
                hout[node * HID + c] = do_relu ? fmaxf(v, 0.0f) : v;
            }
        }
        return;
    }

    // Partial tiles: pre-zero both whole LDS tiles (overlaps the async-written rows below).
    if (!full) {
        for (int i = lane; i < 16 * 16; i += 32) {
            ((float4*)sH)[i] = make_float4(0.f, 0.f, 0.f, 0.f);
            ((float4*)sX)[i] = make_float4(0.f, 0.f, 0.f, 0.f);
        }
    }

    // Stage both 16x64 tiles via async global->LDS (perm-gathered rows).
    uint32_t sHoff = (uint32_t)(uintptr_t)sH;
    uint32_t sXoff = (uint32_t)(uintptr_t)sX;
#pragma unroll
    for (int i = 0; i < 8; ++i) {
        int idx = i * 32 + lane;
        int r = idx >> 4, c4 = idx & 15;
        if (full || (tileIn * 16 + r) < cnt) {
            int node = perm[rowbase + r];
            async_copy_b128(sHoff + (uint32_t)idx * 16, hsum + (size_t)node * HID + c4 * 4);
            async_copy_b128(sXoff + (uint32_t)idx * 16, hin + (size_t)node * HID + c4 * 4);
        }
    }
    wait_asynccnt0();
    __syncthreads();   // single wave: dscnt ordering for the zero-fill path

    int row = lane & 15;
    int khalf = lane >> 4;

    // preload this lane-half's output row indices once
    int nodeOf[8];
#pragma unroll
    for (int r8 = 0; r8 < 8; ++r8) {
        int m = r8 + 8 * khalf;
        nodeOf[r8] = (full || (tileIn * 16 + m) < cnt) ? perm[rowbase + m] : -1;
    }

#pragma unroll
    for (int n0 = 0; n0 < HID; n0 += 16) {
        v8f acc = {};
#pragma unroll
        for (int ks = 0; ks < 16; ++ks) {         // K=64 over hsum x wl
            int kk = ks * 4 + 2 * khalf;
            v2f a; a.x = sH[row * HID + kk]; a.y = sH[row * HID + kk + 1];
            v2f b; b.x = wl[kk * HID + n0 + row]; b.y = wl[(kk + 1) * HID + n0 + row];
            acc = wmma4(a, b, acc);
        }
#pragma unroll
        for (int ks = 0; ks < 16; ++ks) {         // K=64 over hin x wr
            int kk = ks * 4 + 2 * khalf;
            v2f a; a.x = sX[row * HID + kk]; a.y = sX[row * HID + kk + 1];
            v2f b; b.x = wr[kk * HID + n0 + row]; b.y = wr[(kk + 1) * HID + n0 + row];
            acc = wmma4(a, b, acc);
        }
        int col = n0 + row;
        float bb = bl[col];
        if (full) {
#pragma unroll
            for (int r8 = 0; r8 < 8; ++r8) {
                float v = acc[r8] + bb;
                if (do_relu) v = fmaxf(v, 0.0f);
                hout[nodeOf[r8] * HID + col] = v;
            }
        } else {
#pragma unroll
            for (int r8 = 0; r8 < 8; ++r8) {
                if (nodeOf[r8] >= 0) {
                    float v = acc[r8] + bb;
                    if (do_relu) v = fmaxf(v, 0.0f);
                    hout[nodeOf[r8] * HID + col] = v;
                }
            }
        }
    }
}

// ---------------------------------------------------------------- LN + resid
__global__ void k_ln(float* __restrict__ h, const float* __restrict__ h0,
                     const float* __restrict__ g, const float* __restrict__ b, int n) {
    int node = blockIdx.x * (blockDim.x >> 5) + (threadIdx.x >> 5);
    int lane = threadIdx.x & 31;
    if (node >= n) return;
    float v0 = h[node * HID + lane];
    float v1 = h[node * HID + lane + 32];
    float s = v0 + v1;
#pragma unroll
    for (int m = 16; m >= 1; m >>= 1) s += __shfl_xor(s, m, 32);
    float mu = s * (1.0f / HID);
    float d0 = v0 - mu, d1 = v1 - mu;
    float q = d0 * d0 + d1 * d1;
#pragma unroll
    for (int m = 16; m >= 1; m >>= 1) q += __shfl_xor(q, m, 32);
    float rinv = rsqrtf(q * (1.0f / HID) + LN_EPS);
    h[node * HID + lane]      = d0 * rinv * g[lane] + b[lane] + h0[node * HID + lane];
    h[node * HID + lane + 32] = d1 * rinv * g[lane + 32] + b[lane + 32] + h0[node * HID + lane + 32];
}

// ---------------------------------------------------------------- pooling
__global__ void k_pool(const float* __restrict__ h, const unsigned char* __restrict__ ground,
                       const int* __restrict__ batch, float* __restrict__ pooled, int n) {
    int g = blockIdx.x * blockDim.x + threadIdx.x;
    int node = g >> 6, c = g & 63;
    if (node < n && ground[node]) {
        unsafeAtomicAdd(&pooled[batch[node] * HID + c], h[node * HID + c]);
    }
}

__global__ void k_out(const float* __restrict__ pooled, const float* __restrict__ w,
                      const float* __restrict__ b, float* __restrict__ out) {
    int gph = blockIdx.x;
    int lane = threadIdx.x;
    float s = pooled[gph * HID + lane] * w[lane] +
              pooled[gph * HID + lane + 32] * w[lane + 32];
#pragma unroll
    for (int m = 16; m >= 1; m >>= 1) s += __shfl_xor(s, m, 32);
    if (lane == 0) out[gph] = s + b[0];
}

// ---------------------------------------------------------------- host side
extern "C" void kernel_launch(void* const* d_in, const int* in_sizes, int n_in,
                              void* d_out, int out_size, void* d_ws, size_t ws_size,
                              hipStream_t stream) {
    const float* x      = (const float*)d_in[0];
    const int* ei_main  = (const int*)d_in[1];
    const int* ei_sub   = (const int*)d_in[2];
    const int* ei_ns    = (const int*)d_in[3];
    const int* ei_sn    = (const int*)d_in[4];
    const unsigned char* ground = (const unsigned char*)d_in[5];
    const int* batch    = (const int*)d_in[7];
    const float* emb_w  = (const float*)d_in[8];
    const float* emb_b  = (const float*)d_in[9];
    const float* conv_wl = (const float*)d_in[10];   // [4][21][64][64]
    const float* conv_bl = (const float*)d_in[11];   // [4][21][64]
    const float* conv_wr = (const float*)d_in[12];   // [4][21][64][64]
    const float* ln_g   = (const float*)d_in[13];
    const float* ln_b   = (const float*)d_in[14];
    const float* out_w  = (const float*)d_in[15];
    const float* out_b  = (const float*)d_in[16];
    float* out = (float*)d_out;

    char* ws = (char*)d_ws;
    size_t NH = (size_t)N_NODES * HID;
    float* hA   = (float*)ws;                 ws += NH * 4;
    float* hB   = (float*)ws;                 ws += NH * 4;
    float* h0   = (float*)ws;                 ws += NH * 4;
    float* hsum = (float*)ws;                 ws += NH * 4;
    int* deg    = (int*)ws;                   ws += (size_t)N_NODES * 4;
    int* perm   = (int*)ws;                   ws += (size_t)N_NODES * 4;
    int* meta   = (int*)ws;                   ws += 128 * 4;   // 85 used
    float* pooled = (float*)ws;               ws += (size_t)NUM_GRAPHS * HID * 4;

    const int TPB = 256;
    int tilesN = (N_NODES + 15) / 16;

    // embedding (WMMA) -> hA and h0
    k_embed<<<tilesN, 32, 0, stream>>>(x, emb_w, emb_b, hA, h0, N_NODES);

    const int* eis[4] = { ei_main, ei_ns, ei_sub, ei_sn };   // reference order
    float* cur = hA;
    float* nxt = hB;
    for (int j = 0; j < 4; ++j) {
        const int* src = eis[j];
        const int* dst = eis[j] + N_EDGES;
        k_zero<<<(int)((NH + TPB - 1) / TPB), TPB, 0, stream>>>(hsum, (int)NH);
        k_zero_i<<<(N_NODES + TPB - 1) / TPB, TPB, 0, stream>>>(deg, N_NODES);
        k_zero_i<<<1, 32, 0, stream>>>(meta, 21);
        k_deg<<<(N_EDGES + TPB - 1) / TPB, TPB, 0, stream>>>(dst, deg, N_EDGES);
        k_scatter<<<(N_EDGES * HID) / TPB, TPB, 0, stream>>>(src, dst, cur, hsum, N_EDGES);
        k_hist<<<(N_NODES + TPB - 1) / TPB, TPB, 0, stream>>>(deg, meta, N_NODES);
        k_scan<<<1, 32, 0, stream>>>(meta);
        k_permfill<<<(N_NODES + TPB - 1) / TPB, TPB, 0, stream>>>(deg, perm, meta, N_NODES);

        const float* wl = conv_wl + (size_t)j * (MAX_DEG + 1) * HID * HID;
        const float* bl = conv_bl + (size_t)j * (MAX_DEG + 1) * HID;
        const float* wr = conv_wr + (size_t)j * (MAX_DEG + 1) * HID * HID;
        k_conv<<<tilesN + MAX_DEG + 1, 32, 0, stream>>>(hsum, cur, nxt, perm, meta,
                                                        wl, bl, wr, (j < 3) ? 1 : 0);
        float* t = cur; cur = nxt; nxt = t;
    }

    // layernorm + residual (in place on cur)
    k_ln<<<(N_NODES + 7) / 8, 256, 0, stream>>>(cur, h0, ln_g, ln_b, N_NODES);

    // pool over ground nodes, then final matvec
    k_zero<<<(NUM_GRAPHS * HID + TPB - 1) / TPB, TPB, 0, stream>>>(pooled, NUM_GRAPHS * HID);
    k_pool<<<(N_NODES * HID) / TPB, TPB, 0, stream>>>(cur, ground, batch, pooled, N_NODES);
    k_out<<<NUM_GRAPHS, 32, 0, stream>>>(pooled, out_w, out_b, out);
}